// GMMEmbedder_80625126081173
// MI455X (gfx1250) — compile-verified
//
#include <hip/hip_runtime.h>
#include <hip/hip_bf16.h>
#include <math.h>

// Problem constants (from reference): K=8, D=4, M=512, B=4096, phi_dim=120.
#define MM 512
#define KK 8
#define DDIM 4
#define BB 4096
#define PHI_DIM 120
#define LOG2PI 1.8378770664093453f

typedef __attribute__((ext_vector_type(2))) float v2f;
typedef __attribute__((ext_vector_type(8))) float v8f;

// Workspace layout (floats):
//   W : [4096][16]      per-(m,k) coefficient vectors (8 x v2f per row)
//   Yp: [8][4096][2]    monomial features packed as K-row pairs (v2f)
//   R : [4096]          resp accumulators (sum over b)
//   Z : [512][80]       un-normalized embedding
#define WS_W 0
#define WS_Y 65536
#define WS_R 131072
#define WS_Z 135168

// ---------------------------------------------------------------------------
// Kernel 1: per-(m,k) precompute.  maha = x^T P x - 2 g.x + c0 with
// P = Linv^T Linv.  Fold everything (log pi, log det, D*log2pi) into w[16] so
// log_joint(m,k,b) = -0.5 * dot(w_mk, y_b).  Also zeroes the resp accumulator.
// ---------------------------------------------------------------------------
__global__ __launch_bounds__(256) void gmm_prep(const float* __restrict__ phi,
                                                float* __restrict__ W,
                                                float* __restrict__ R) {
  int t = blockIdx.x * blockDim.x + threadIdx.x;
  if (t >= MM * KK) return;
  int m = t >> 3;
  int k = t & 7;
  const float* row = phi + m * PHI_DIM;

  // softmax over pi_tilde (8 values)
  float mx = row[0];
  #pragma unroll
  for (int i = 1; i < 8; ++i) mx = fmaxf(mx, row[i]);
  float se = 0.f;
  #pragma unroll
  for (int i = 0; i < 8; ++i) se += expf(row[i] - mx);
  float pi = expf(row[k] - mx) / se;

  float mu[4];
  #pragma unroll
  for (int i = 0; i < 4; ++i) mu[i] = row[8 + k * 4 + i];

  const float* Lv = row + 40 + k * 10;  // tril order (0,0)(1,0)(1,1)(2,0)(2,1)(2,2)(3,0)(3,1)(3,2)(3,3)
  float L[4][4] = {{Lv[0], 0.f, 0.f, 0.f},
                   {Lv[1], Lv[2], 0.f, 0.f},
                   {Lv[3], Lv[4], Lv[5], 0.f},
                   {Lv[6], Lv[7], Lv[8], Lv[9]}};

  // Invert lower-triangular L by forward substitution (raw diag, matching the
  // reference's solve_triangular).
  float Li[4][4] = {};
  #pragma unroll
  for (int i = 0; i < 4; ++i) {
    Li[i][i] = 1.0f / L[i][i];
    #pragma unroll
    for (int j = 0; j < 4; ++j) {
      if (j < i) {
        float s = 0.f;
        for (int tt = j; tt < i; ++tt) s += L[i][tt] * Li[tt][j];
        Li[i][j] = -s / L[i][i];
      }
    }
  }

  // P = Linv^T Linv (symmetric precision matrix)
  float P[4][4];
  #pragma unroll
  for (int i = 0; i < 4; ++i)
    #pragma unroll
    for (int j = 0; j < 4; ++j) {
      float s = 0.f;
      int lo = i > j ? i : j;
      for (int tt = lo; tt < 4; ++tt) s += Li[tt][i] * Li[tt][j];
      P[i][j] = s;
    }

  float g[4];
  #pragma unroll
  for (int i = 0; i < 4; ++i) {
    float s = 0.f;
    #pragma unroll
    for (int j = 0; j < 4; ++j) s += P[i][j] * mu[j];
    g[i] = s;
  }
  float c0 = 0.f;
  #pragma unroll
  for (int i = 0; i < 4; ++i) c0 += g[i] * mu[i];

  float ld = 0.f;
  #pragma unroll
  for (int i = 0; i < 4; ++i) ld += logf(fmaxf(fabsf(L[i][i]), 1e-8f));
  ld *= 2.0f;

  // S = w.y ; log_joint = -0.5 * S
  float cst = c0 + (float)DDIM * LOG2PI + ld - 2.0f * logf(fmaxf(pi, 1e-8f));

  float* w = W + t * 16;
  w[0] = P[0][0]; w[1] = P[1][1]; w[2] = P[2][2]; w[3] = P[3][3];
  w[4] = 2.f * P[0][1]; w[5] = 2.f * P[0][2]; w[6] = 2.f * P[0][3];
  w[7] = 2.f * P[1][2]; w[8] = 2.f * P[1][3]; w[9] = 2.f * P[2][3];
  w[10] = -2.f * g[0]; w[11] = -2.f * g[1]; w[12] = -2.f * g[2]; w[13] = -2.f * g[3];
  w[14] = cst;
  w[15] = 0.f;

  R[t] = 0.f;  // zero resp accumulator (same index space: 4096)
}

// ---------------------------------------------------------------------------
// Kernel 2: monomial features per reference point b, packed as K-row pairs so
// each WMMA B-fragment is a single aligned b64 load:
//   Yp[p][b] = { y[2p][b], y[2p+1][b] },  p = 0..7
// y = [x0^2,x1^2,x2^2,x3^2, x0x1,x0x2,x0x3,x1x2,x1x3,x2x3, x0,x1,x2,x3, 1, 0]
// ---------------------------------------------------------------------------
__global__ __launch_bounds__(256) void gmm_features(const float* __restrict__ X,
                                                    v2f* __restrict__ Yp) {
  int b = blockIdx.x * blockDim.x + threadIdx.x;
  if (b >= BB) return;
  float x0 = X[b * 4 + 0], x1 = X[b * 4 + 1], x2 = X[b * 4 + 2], x3 = X[b * 4 + 3];
  Yp[0 * BB + b] = (v2f){x0 * x0, x1 * x1};
  Yp[1 * BB + b] = (v2f){x2 * x2, x3 * x3};
  Yp[2 * BB + b] = (v2f){x0 * x1, x0 * x2};
  Yp[3 * BB + b] = (v2f){x0 * x3, x1 * x2};
  Yp[4 * BB + b] = (v2f){x1 * x3, x2 * x3};
  Yp[5 * BB + b] = (v2f){x0, x1};
  Yp[6 * BB + b] = (v2f){x2, x3};
  Yp[7 * BB + b] = (v2f){1.0f, 0.0f};
}

// ---------------------------------------------------------------------------
// Kernel 3: the GEMM.  Output rows = flattened (m,k) [4096], cols = b [4096],
// reduction dim = 16 -> 4 chained V_WMMA_F32_16X16X4_F32 per 16x16 tile.
// One wave per row-tile x 512-column strip; A-fragments loaded once and
// reused across 32 column tiles.  WMMA C-layout puts all 8 k-values for one
// (m,b) in one lane's 8 accumulator regs -> softmax over K is lane-local:
// 8 x v_exp_f32 + 1 v_rcp_f32 per tile (exp(t-lse) == exp(t-mx)/sum).
// ---------------------------------------------------------------------------
__global__ __launch_bounds__(256) void gmm_maha_resp(const float* __restrict__ W,
                                                     const float* __restrict__ Y,
                                                     float* __restrict__ R) {
  const int lane  = threadIdx.x & 31;
  const int wave  = threadIdx.x >> 5;   // 0..7 -> 512-column strip
  const int tr    = blockIdx.x;         // 0..255 row tile (covers m=2*tr, 2*tr+1)
  const int laneM = lane & 15;
  const int kpair = lane >> 4;          // 0: K=0,1 ; 1: K=2,3 within each K-chunk

  const v2f* Wp = (const v2f*)W;        // [4096][8] pairs
  const v2f* Yp = (const v2f*)Y;        // [8][4096] pairs

  // A fragments: row = tr*16 + laneM; chunk c uses pairs {2c+kpair}
  v2f a[4];
  const int rowA = tr * 16 + laneM;
  #pragma unroll
  for (int c = 0; c < 4; ++c) a[c] = Wp[rowA * 8 + 2 * c + kpair];

  float racc[8];
  #pragma unroll
  for (int k = 0; k < 8; ++k) racc[k] = 0.f;

  const int colStrip = wave * 512;
  for (int ct = 0; ct < 32; ++ct) {
    const int col = colStrip + ct * 16 + laneM;
    v8f acc = {};
    #pragma unroll
    for (int c = 0; c < 4; ++c) {
      v2f bf = Yp[(2 * c + kpair) * BB + col];
      acc = __builtin_amdgcn_wmma_f32_16x16x4_f32(
          /*neg_a=*/false, a[c], /*neg_b=*/false, bf,
          /*c_mod=*/(short)0, acc, /*reuse_a=*/false, /*reuse_b=*/false);
    }
    // lane-local: this lane holds S for k=0..7 of one (m,b). softmax over K.
    float t[8];
    #pragma unroll
    for (int k = 0; k < 8; ++k) t[k] = -0.5f * acc[k];
    float mx = t[0];
    #pragma unroll
    for (int k = 1; k < 8; ++k) mx = fmaxf(mx, t[k]);
    float e[8];
    float s = 0.f;
    #pragma unroll
    for (int k = 0; k < 8; ++k) { e[k] = __expf(t[k] - mx); s += e[k]; }
    // softmax denominator is in [1,8]: hardware v_rcp_f32 (<=1 ulp) suffices,
    // avoids the full IEEE div_scale/div_fmas/div_fixup chain in the hot loop.
    const float inv = __builtin_amdgcn_rcpf(s);
    #pragma unroll
    for (int k = 0; k < 8; ++k) racc[k] += e[k] * inv;
  }

  // Reduce over the 16 lanes sharing the same m (xor masks < 16 never cross
  // the half-wave boundary), then one atomic per (m,k).
  #pragma unroll
  for (int k = 0; k < 8; ++k) {
    float v = racc[k];
    v += __shfl_xor(v, 1);
    v += __shfl_xor(v, 2);
    v += __shfl_xor(v, 4);
    v += __shfl_xor(v, 8);
    racc[k] = v;
  }
  if (laneM == 0) {
    const int m = 2 * tr + kpair;
    #pragma unroll
    for (int k = 0; k < 8; ++k) atomicAdd(&R[m * 8 + k], racc[k]);
  }
}

// ---------------------------------------------------------------------------
// Kernel 4: assemble un-normalized z[m][80] = [mu(32) | log|diagL|(32) | pi(8) | resp(8)]
// ---------------------------------------------------------------------------
__global__ __launch_bounds__(256) void gmm_z(const float* __restrict__ phi,
                                             const float* __restrict__ R,
                                             float* __restrict__ Z) {
  int m = blockIdx.x * blockDim.x + threadIdx.x;
  if (m >= MM) return;
  const float* row = phi + m * PHI_DIM;
  float* z = Z + m * 80;
  #pragma unroll
  for (int i = 0; i < 32; ++i) z[i] = row[8 + i];
  const int dpos[4] = {0, 2, 5, 9};  // diagonal positions within the 10-vec
  #pragma unroll
  for (int k = 0; k < 8; ++k)
    #pragma unroll
    for (int d = 0; d < 4; ++d)
      z[32 + k * 4 + d] = logf(fmaxf(fabsf(row[40 + k * 10 + dpos[d]]), 1e-6f));
  float mx = row[0];
  #pragma unroll
  for (int i = 1; i < 8; ++i) mx = fmaxf(mx, row[i]);
  float se = 0.f;
  #pragma unroll
  for (int i = 0; i < 8; ++i) se += expf(row[i] - mx);
  #pragma unroll
  for (int k = 0; k < 8; ++k) z[64 + k] = expf(row[k] - mx) / se;
  #pragma unroll
  for (int k = 0; k < 8; ++k) z[72 + k] = R[m * 8 + k] * (1.0f / (float)BB);
}

// ---------------------------------------------------------------------------
// Kernel 5: column standardization over M=512 rows (ddof=1, std clipped 1e-6).
// One block per column.
// ---------------------------------------------------------------------------
__global__ __launch_bounds__(256) void gmm_norm(const float* __restrict__ Z,
                                                float* __restrict__ out) {
  __shared__ float ssum[256];
  __shared__ float ssq[256];
  const int j = blockIdx.x;   // 0..79
  const int t = threadIdx.x;  // handles rows t and t+256
  float v0 = Z[t * 80 + j];
  float v1 = Z[(t + 256) * 80 + j];
  ssum[t] = v0 + v1;
  ssq[t]  = v0 * v0 + v1 * v1;
  __syncthreads();
  for (int off = 128; off > 0; off >>= 1) {
    if (t < off) { ssum[t] += ssum[t + off]; ssq[t] += ssq[t + off]; }
    __syncthreads();
  }
  const float sum = ssum[0];
  const float mean = sum / 512.0f;
  float var = (ssq[0] - sum * sum / 512.0f) / 511.0f;
  float sd = sqrtf(fmaxf(var, 0.0f));
  float denom = fmaxf(sd, 1e-6f);
  out[t * 80 + j] = (v0 - mean) / denom;
  out[(t + 256) * 80 + j] = (v1 - mean) / denom;
}

// ---------------------------------------------------------------------------
extern "C" void kernel_launch(void* const* d_in, const int* in_sizes, int n_in,
                              void* d_out, int out_size, void* d_ws, size_t ws_size,
                              hipStream_t stream) {
  (void)in_sizes; (void)n_in; (void)out_size; (void)ws_size;
  const float* phi = (const float*)d_in[0];   // (512, 120) f32
  const float* X   = (const float*)d_in[1];   // (4096, 4) f32
  float* out = (float*)d_out;                 // (512, 80) f32
  float* ws = (float*)d_ws;
  float* W  = ws + WS_W;
  float* Yf = ws + WS_Y;
  float* R  = ws + WS_R;
  float* Z  = ws + WS_Z;

  gmm_prep<<<(MM * KK + 255) / 256, 256, 0, stream>>>(phi, W, R);
  gmm_features<<<(BB + 255) / 256, 256, 0, stream>>>(X, (v2f*)Yf);
  gmm_maha_resp<<<256, 256, 0, stream>>>(W, Yf, R);
  gmm_z<<<(MM + 255) / 256, 256, 0, stream>>>(phi, R, Z);
  gmm_norm<<<80, 256, 0, stream>>>(Z, out);
}